// TSCTranspose_input_18751827214349
// MI455X (gfx1250) — compile-verified
//
#include <hip/hip_runtime.h>

#define OUT_LEN 16384
#define B_ROWS  2048
#define ENC_LEN 16368   // sum_{e=1..10} 16384>>e
#define WLEN    2047

typedef unsigned int v4u __attribute__((ext_vector_type(4)));
typedef int          v8i __attribute__((ext_vector_type(8)));
typedef int          v4i __attribute__((ext_vector_type(4)));

// Low 32 bits of a generic (flat) pointer into LDS == LDS byte offset
// (ISA §10.2: LDS aperture maps addr[31:0] directly).
__device__ __forceinline__ unsigned lds_offset_of(const void* p) {
    return (unsigned)(unsigned long long)(uintptr_t)p;
}

// Issue one 1-D TDM load: nelem fp32 from gptr -> LDS at lds_off.
// D# layout per CDNA5 ISA ch.8 (groups 0/1; groups 2/3 zero for <=2D).
// This toolchain's builtin is the 6-arg form:
//   (u32x4 g0, i32x8 g1, i32x4 g2, i32x4 g3, i32x8 extra, i32 cpol)
__device__ __forceinline__ void tdm_load_row(const float* gptr, unsigned lds_off, int nelem) {
    unsigned long long ga = (unsigned long long)(uintptr_t)gptr;
    v4u g0;
    g0.x = 1u;                                           // count=1 (valid), user mode
    g0.y = lds_off;                                      // lds_addr (bytes)
    g0.z = (unsigned)(ga & 0xFFFFFFFFull);               // global_addr[31:0]
    g0.w = (unsigned)((ga >> 32) & 0x01FFFFFFull)        // global_addr[56:32]
         | (2u << 30);                                   // type=2 ("image")
    v8i g1;
    g1[0] = 2 << 16;                                     // data_size=2 (4 bytes)
    g1[1] = (int)((unsigned)(nelem & 0xFFFF) << 16);     // tensor_dim0[15:0]
    g1[2] = ((nelem >> 16) & 0xFFFF) | (1 << 16);        // tensor_dim0[31:16] | tensor_dim1=1
    g1[3] = (int)((unsigned)(nelem & 0xFFFF) << 16);     // tile_dim0 = nelem (16-bit)
    g1[4] = 1;                                           // tile_dim1=1, tile_dim2=0
    g1[5] = nelem;                                       // tensor_dim0_stride[31:0]
    g1[6] = 0;                                           // stride hi, tensor_dim1_stride lo
    g1[7] = 0;
    v4i z4 = {0, 0, 0, 0};
    v8i z8 = {0, 0, 0, 0, 0, 0, 0, 0};
    __builtin_amdgcn_tensor_load_to_lds(g0, g1, z4, z4, z8, 0);
}

__global__ __launch_bounds__(256) void tsc_transpose_kernel(
    const float* __restrict__ data,     // (2048, 16368)
    const float* __restrict__ weights,  // (2047,)
    const float* __restrict__ bias,     // (10,)
    float* __restrict__ out)            // (2048, 16384)
{
    __shared__ float s_data[ENC_LEN];
    __shared__ float s_w[2048];

    const int t = threadIdx.x;
    const int b = blockIdx.x;
    const int j0 = 4 * t;               // first of this thread's 4 outputs within a 1024-chunk

    // --- Stage row + weight table into LDS via the Tensor Data Mover. ---
    // TDM ignores EXEC; the wave-0 branch prevents 8 duplicate DMAs.
    if (t < 32) {
        tdm_load_row(data + (size_t)b * ENC_LEN, lds_offset_of(&s_data[0]), ENC_LEN);
        tdm_load_row(weights, lds_offset_of(&s_w[0]), WLEN);
        __builtin_amdgcn_s_wait_tensorcnt(0);   // TENSORcnt==0: both DMAs complete
    }
    __syncthreads();                            // release remaining 7 waves

    // --- Per-thread invariant weights: j mod 2^e is independent of k since 1024 % 2^e == 0. ---
    float Wv[10][4];
#pragma unroll
    for (int e = 1; e <= 10; ++e) {
        const int w = 1 << e;
#pragma unroll
        for (int i = 0; i < 4; ++i)
            Wv[e - 1][i] = s_w[(w - 1) + ((j0 + i) & (w - 1))];
    }

    float bsum = 0.0f;
#pragma unroll
    for (int i = 0; i < 10; ++i) bsum += bias[i];   // uniform: scalar loads, L2-hot

    // --- Per-scale running LDS indices; advance by (1024>>e) per k-chunk. ---
    int idx[10];
    {
        int off = 0;
#pragma unroll
        for (int e = 1; e <= 10; ++e) {
            const int c = (e == 1) ? (2 * t) : (j0 >> e);
            idx[e - 1] = off + c;
            off += OUT_LEN >> e;
        }
    }

    float* outp = out + (size_t)b * OUT_LEN + j0;

#pragma unroll 1
    for (int k = 0; k < 16; ++k) {
        float4 acc = make_float4(bsum, bsum, bsum, bsum);

        // e = 1: two data values cover the 4 outputs (b64 LDS load, fully coalesced)
        const float2 d1 = *reinterpret_cast<const float2*>(&s_data[idx[0]]);
        acc.x += d1.x * Wv[0][0];
        acc.y += d1.x * Wv[0][1];
        acc.z += d1.y * Wv[0][2];
        acc.w += d1.y * Wv[0][3];

        // e = 2..10: one shared data value per thread (broadcast-friendly LDS reads)
#pragma unroll
        for (int e = 2; e <= 10; ++e) {
            const float d = s_data[idx[e - 1]];
            acc.x += d * Wv[e - 1][0];
            acc.y += d * Wv[e - 1][1];
            acc.z += d * Wv[e - 1][2];
            acc.w += d * Wv[e - 1][3];
        }

        *reinterpret_cast<float4*>(outp + (k << 10)) = acc;   // 128-bit coalesced store

#pragma unroll
        for (int e = 1; e <= 10; ++e) idx[e - 1] += (1024 >> e);
    }
}

extern "C" void kernel_launch(void* const* d_in, const int* in_sizes, int n_in,
                              void* d_out, int out_size, void* d_ws, size_t ws_size,
                              hipStream_t stream) {
    const float* data    = (const float*)d_in[0];
    const float* weights = (const float*)d_in[1];
    const float* bias    = (const float*)d_in[2];
    float* out = (float*)d_out;

    dim3 grid(B_ROWS);
    dim3 block(256);
    tsc_transpose_kernel<<<grid, block, 0, stream>>>(data, weights, bias, out);
}